// StructureLoss_5884105196151
// MI455X (gfx1250) — compile-verified
//
#include <hip/hip_runtime.h>
#include <math.h>

// ---------------------------------------------------------------------------
// StructureLoss for MI455X (gfx1250):
//   31x31 box filter (separable) + weighted BCE + weighted IoU -> scalar.
//   Stage 1: global -> LDS via gfx1250 async-to-LDS path (ASYNCcnt) if the
//            toolchain exposes the builtin, else load/store fallback.
//   Stage 2: horizontal 31-tap via sliding-window running sums in LDS.
//   Stage 3: vertical 31-tap as banded GEMM on the WMMA pipe:
//            O[16x16] = Band[16x48] x In[48x16], 12x v_wmma_f32_16x16x4_f32.
//   Stage 4: pointwise BCE/IoU terms + deterministic hierarchical reduction.
// ---------------------------------------------------------------------------

typedef __attribute__((ext_vector_type(2))) float v2f;
typedef __attribute__((ext_vector_type(8))) float v8f;

#define IMG_H   512
#define IMG_W   512
#define NIMG    32
#define KF      31
#define PADF    15
#define TILE_R  16
#define TILE_C  128                 // 8 waves x 16 cols
#define IN_R    48                  // TILE_R + 2*PADF (=46) padded to K=48
#define IN_C    (TILE_C + KF - 1)   // 158
#define IN_CP   160                 // padded LDS stride
#define HS_CP   132                 // padded LDS stride for row-filtered tile
#define RT_PER_IMG (IMG_H / TILE_R) // 32
#define CT_PER_IMG (IMG_W / TILE_C) // 4
#define BLK_PER_IMG (RT_PER_IMG * CT_PER_IMG) // 128
#define NBLK (NIMG * BLK_PER_IMG)             // 4096

#if defined(__has_builtin)
#if __has_builtin(__builtin_amdgcn_global_load_async_to_lds_b32) && \
    __has_builtin(__builtin_amdgcn_s_wait_asynccnt)
#define HAVE_ASYNC_LDS 1
#endif
#endif
#ifndef HAVE_ASYNC_LDS
#define HAVE_ASYNC_LDS 0
#endif

#if HAVE_ASYNC_LDS
typedef __attribute__((address_space(1))) int* gptr_t;
typedef __attribute__((address_space(3))) int* lptr_t;
#endif

__global__ __launch_bounds__(256)
void structure_loss_main(const float* __restrict__ pred,
                         const float* __restrict__ target,
                         float* __restrict__ partials)
{
    __shared__ float tin[IN_R][IN_CP];   // raw target tile (with halos), ~30.7 KB
    __shared__ float hs[IN_R][HS_CP];    // horizontally box-summed tile, ~25.3 KB
    __shared__ float red[8][4];          // per-wave partials

    const int tid = threadIdx.x;
    const int blk = blockIdx.x;
    const int n   = blk >> 7;                      // / BLK_PER_IMG
    const int rem = blk & (BLK_PER_IMG - 1);
    const int r0  = (rem >> 2) * TILE_R;           // 32 row tiles
    const int c0  = (rem & 3) * TILE_C;            // 4 col tiles

    const float* timg = target + (size_t)n * IMG_H * IMG_W;
    const float* pimg = pred   + (size_t)n * IMG_H * IMG_W;

    // gfx1250 prefetch of the pred tile we will need in stage 4
    __builtin_prefetch(pimg + (size_t)r0 * IMG_W + c0, 0, 3);

    // ---- stage 1: target halo tile (48 x 158) -> LDS, zero-padded ----
    for (int idx = tid; idx < IN_R * IN_C; idx += 256) {
        int lr = idx / IN_C;
        int lc = idx - lr * IN_C;
        int gr = r0 - PADF + lr;
        int gc = c0 - PADF + lc;
        const bool inr = (gr >= 0 && gr < IMG_H && gc >= 0 && gc < IMG_W);
#if HAVE_ASYNC_LDS
        if (inr) {
            // global_load_async_to_lds_b32: LDS written directly, ASYNCcnt-tracked
            __builtin_amdgcn_global_load_async_to_lds_b32(
                (gptr_t)(void*)(timg + (size_t)gr * IMG_W + gc),
                (lptr_t)(void*)&tin[lr][lc],
                0, 0);
        } else {
            tin[lr][lc] = 0.0f;
        }
#else
        tin[lr][lc] = inr ? timg[(size_t)gr * IMG_W + gc] : 0.0f;
#endif
    }
#if HAVE_ASYNC_LDS
    __builtin_amdgcn_s_wait_asynccnt(0);
#endif
    __syncthreads();

    // ---- stage 2: horizontal 31-tap sums via sliding window ----
    // 48 rows x 4 segments of 32 cols = 192 tasks; running sum per segment.
    if (tid < IN_R * 4) {
        const int lr = tid >> 2;          // 0..47
        const int j0 = (tid & 3) * 32;    // segment start
        float s = 0.0f;
        #pragma unroll
        for (int d = 0; d < KF; ++d) s += tin[lr][j0 + d];
        hs[lr][j0] = s;
        #pragma unroll 4
        for (int j = j0 + 1; j < j0 + 32; ++j) {
            s += tin[lr][j + (KF - 1)] - tin[lr][j - 1];
            hs[lr][j] = s;
        }
    }
    __syncthreads();

    // ---- stage 3: vertical 31-tap as banded GEMM on the WMMA pipe ----
    // O[16x16] = Band[16x48] * hs[48x16];  Band[i][k] = (i <= k <= i+30)
    const int lane = tid & 31;
    const int w    = tid >> 5;     // wave id 0..7 -> 16-col slice
    const int half = lane >> 4;    // K split across lane halves
    const int m    = lane & 15;    // M for A, N for B/C/D

    v8f acc = {};                  // C = 0
    #pragma unroll
    for (int kk = 0; kk < 12; ++kk) {
        const int k0 = kk * 4 + half * 2;
        v2f a, b;
        a[0] = (k0     >= m && k0     <= m + 30) ? 1.0f : 0.0f;
        a[1] = (k0 + 1 >= m && k0 + 1 <= m + 30) ? 1.0f : 0.0f;
        b[0] = hs[k0    ][w * 16 + m];
        b[1] = hs[k0 + 1][w * 16 + m];
        // D = A x B + C ; emits v_wmma_f32_16x16x4_f32
        acc = __builtin_amdgcn_wmma_f32_16x16x4_f32(
                  false, a, false, b, (short)0, acc, false, false);
    }

    // ---- stage 4: pointwise loss terms + per-thread accumulation ----
    // acc[v] holds box-sum S at tile row (v + 8*half), tile col (16w + m)
    float s_wsum = 0.0f, s_wbce = 0.0f, s_inter = 0.0f, s_union = 0.0f;
    const float inv_kk = 1.0f / (float)(KF * KF);   // 1/961
    #pragma unroll
    for (int v = 0; v < 8; ++v) {
        const int rr = v + half * 8;                // 0..15
        const int cc = w * 16 + m;                  // 0..127
        const float t   = tin[rr + PADF][cc + PADF];
        const float x   = pimg[(r0 + rr) * IMG_W + (c0 + cc)];
        const float avg = acc[v] * inv_kk;
        const float weit = 1.0f + 5.0f * fabsf(avg - t);
        // stable BCE-with-logits
        const float wb = fmaxf(x, 0.0f) - x * t + log1pf(expf(-fabsf(x)));
        const float p  = 1.0f / (1.0f + expf(-x));
        s_wsum  += weit;
        s_wbce  += weit * wb;
        s_inter += p * t * weit;
        s_union += (p + t) * weit;
    }

    // wave32 reduction
    #pragma unroll
    for (int off = 16; off > 0; off >>= 1) {
        s_wsum  += __shfl_xor(s_wsum,  off, 32);
        s_wbce  += __shfl_xor(s_wbce,  off, 32);
        s_inter += __shfl_xor(s_inter, off, 32);
        s_union += __shfl_xor(s_union, off, 32);
    }
    if (lane == 0) {
        red[w][0] = s_wsum; red[w][1] = s_wbce;
        red[w][2] = s_inter; red[w][3] = s_union;
    }
    __syncthreads();
    if (tid < 4) {
        float v4 = 0.0f;
        #pragma unroll
        for (int i = 0; i < 8; ++i) v4 += red[i][tid];
        partials[blk * 4 + tid] = v4;   // deterministic (no atomics)
    }
}

// ---- per-image combine: 128 block-partials -> per-image loss value ----
__global__ __launch_bounds__(128)
void structure_loss_img(const float* __restrict__ partials,
                        float* __restrict__ per_img)
{
    __shared__ float sm[128][4];
    const int n = blockIdx.x;
    const int t = threadIdx.x;
    const float* p = partials + ((size_t)n * BLK_PER_IMG + t) * 4;
    sm[t][0] = p[0]; sm[t][1] = p[1]; sm[t][2] = p[2]; sm[t][3] = p[3];
    __syncthreads();
    for (int off = 64; off > 0; off >>= 1) {
        if (t < off) {
            sm[t][0] += sm[t + off][0];
            sm[t][1] += sm[t + off][1];
            sm[t][2] += sm[t + off][2];
            sm[t][3] += sm[t + off][3];
        }
        __syncthreads();
    }
    if (t == 0) {
        const float wsum  = sm[0][0];
        const float wbceS = sm[0][1];
        const float inter = sm[0][2];
        const float uni   = sm[0][3];
        const float wbce  = wbceS / wsum;
        const float wiou  = 1.0f - (inter + 1.0f) / (uni - inter + 1.0f);
        per_img[n] = wbce + wiou;
    }
}

// ---- final mean over 32 images -> scalar ----
__global__ void structure_loss_final(const float* __restrict__ per_img,
                                     float* __restrict__ out)
{
    float v = per_img[threadIdx.x];
    #pragma unroll
    for (int off = 16; off > 0; off >>= 1) v += __shfl_xor(v, off, 32);
    if (threadIdx.x == 0) out[0] = v * (1.0f / (float)NIMG);
}

extern "C" void kernel_launch(void* const* d_in, const int* in_sizes, int n_in,
                              void* d_out, int out_size, void* d_ws, size_t ws_size,
                              hipStream_t stream)
{
    const float* pred   = (const float*)d_in[0];
    const float* target = (const float*)d_in[1];
    float* ws       = (float*)d_ws;
    float* partials = ws;                    // NBLK*4 floats = 64 KB
    float* per_img  = ws + (size_t)NBLK * 4; // 32 floats

    structure_loss_main<<<NBLK, 256, 0, stream>>>(pred, target, partials);
    structure_loss_img<<<NIMG, 128, 0, stream>>>(partials, per_img);
    structure_loss_final<<<1, 32, 0, stream>>>(per_img, (float*)d_out);
}